// GATLayer_84404697301688
// MI455X (gfx1250) — compile-verified
//
#include <hip/hip_runtime.h>
#include <math.h>

#define BB 4
#define NN 2048
#define KIN 128
#define HH 4
#define DD 64
#define HD 256            // HH*DD
#define ROWS (BB * NN)    // 8192
#define NBH  (BB * HH)    // 16
#define LEAKY 0.2f
#define EPS_BN 1e-5f

typedef __attribute__((ext_vector_type(2))) float v2f;
typedef __attribute__((ext_vector_type(8))) float v8f;
typedef __attribute__((ext_vector_type(4))) unsigned int u32x4;
typedef __attribute__((ext_vector_type(8))) int i32x8;
typedef __attribute__((ext_vector_type(4))) int i32x4;

// workspace layout (float offsets)
// WhP2: K-pair interleaved Wh:  WhP2[bh][n/2][c][2], c=0..63, total ROWS*HD
#define OFF_WH   0
#define OFF_SRC  (OFF_WH + (size_t)ROWS * HD)          // 2,097,152
#define OFF_DST  (OFF_SRC + (size_t)NBH * NN)          // +32768
#define OFF_M    (OFF_DST + (size_t)NBH * NN)
#define OFF_L    (OFF_M + (size_t)NBH * NN)
#define OFF_S1   (OFF_L + (size_t)NBH * NN)            // 256 (mean)
#define OFF_S2   (OFF_S1 + 256)                        // 256 (inv-std)
#define OFF_P1   (OFF_S2 + 256)                        // 64*256 partial sums
#define OFF_P2   (OFF_P1 + 64 * 256)                   // 64*256 partial sumsq

// ---------------------------------------------------------------------------
// TDM: contiguous copy of nElem 4-byte words from global to LDS.
// D# built per CDNA5 ISA 8.3/8.4: group0 = {count|..., lds_addr,
// global_addr[31:0], global_addr[56:32] | type=2<<30}; group1 packs
// data_size=4B, tensor_dim0=nElem, tensor_dim1=1, tile_dim0=nElem (<=65535),
// tile_dim1=1, stride0=nElem. Groups 2/3 unused (<=2D).
// ---------------------------------------------------------------------------
__device__ __forceinline__ void tdm_copy_to_lds(unsigned int lds_byte_off,
                                                const float* gsrc,
                                                unsigned int nElem) {
    unsigned long long ga = (unsigned long long)(uintptr_t)gsrc;
    u32x4 g0;
    g0.x = 1u;                                              // count=1 (valid)
    g0.y = lds_byte_off;                                    // lds_addr
    g0.z = (unsigned int)ga;                                // global_addr lo
    g0.w = (unsigned int)((ga >> 32) & 0x1FFFFFFu) | 0x80000000u; // hi | type=2
    i32x8 g1;
    g1[0] = 0x00020000;                                     // data_size = 4B
    g1[1] = (int)(nElem << 16);                             // tensor_dim0[15:0]
    g1[2] = (int)((nElem >> 16) | (1u << 16));              // td0 hi | td1=1
    g1[3] = (int)(nElem << 16);                             // tile_dim0
    g1[4] = 1;                                              // tile_dim1=1
    g1[5] = (int)nElem;                                     // stride0 lo
    g1[6] = 0;
    g1[7] = 0;
    i32x4 z4 = {0, 0, 0, 0};
    i32x8 z8 = {0, 0, 0, 0, 0, 0, 0, 0};
    __builtin_amdgcn_tensor_load_to_lds(g0, g1, z4, z4, z8, 0);
}

// ---------------------------------------------------------------------------
// Kernel 1: Wh = h @ W  (M=8192, K=128, N=256) via V_WMMA_F32_16X16X4_F32.
// One wave per 16x16 tile; output written in K-pair-interleaved WhP2 layout.
// ---------------------------------------------------------------------------
__global__ __launch_bounds__(128) void k_gemm_wh(const float* __restrict__ h,
                                                 const float* __restrict__ W,
                                                 float* __restrict__ WhP2) {
    int wave = (int)((blockIdx.x * blockDim.x + threadIdx.x) >> 5);
    int lane = threadIdx.x & 31;
    int half = lane >> 4;
    int l    = lane & 15;
    int trow = wave >> 4;     // 0..511
    int tcol = wave & 15;     // 0..15
    int row0 = trow * 16, col0 = tcol * 16;

    v8f c = {0.f, 0.f, 0.f, 0.f, 0.f, 0.f, 0.f, 0.f};
    const float* hrow = h + (size_t)(row0 + l) * KIN;
    for (int k0 = 0; k0 < KIN; k0 += 4) {
        int ka = k0 + 2 * half;
        v2f av, bv;
        av.x = hrow[ka];
        av.y = hrow[ka + 1];
        bv.x = W[(size_t)ka * HD + col0 + l];
        bv.y = W[(size_t)(ka + 1) * HD + col0 + l];
        c = __builtin_amdgcn_wmma_f32_16x16x4_f32(false, av, false, bv,
                                                  (short)0, c, false, false);
    }
    int col = col0 + l;
    int hh  = col >> 6;
    int cc  = col & 63;
#pragma unroll
    for (int r = 0; r < 8; ++r) {
        int row  = row0 + r + 8 * half;
        int bIdx = row >> 11;
        int n    = row & (NN - 1);
        size_t pr = (size_t)((bIdx * HH + hh) * (NN / 2) + (n >> 1));
        WhP2[pr * 128 + cc * 2 + (n & 1)] = c[r];
    }
}

// ---------------------------------------------------------------------------
// Kernel 2: src/dst per-row attention dot products (from WhP2 layout).
// ---------------------------------------------------------------------------
__global__ __launch_bounds__(128) void k_srcdst(const float* __restrict__ WhP2,
                                                const float* __restrict__ a,
                                                float* __restrict__ src,
                                                float* __restrict__ dst) {
    int wave = (int)((blockIdx.x * blockDim.x + threadIdx.x) >> 5); // bh*NN+n
    int lane = threadIdx.x & 31;
    int bh = wave / NN;
    int n  = wave % NN;
    int hh = bh % HH;
    const float* wb = WhP2 + (size_t)(bh * (NN / 2) + (n >> 1)) * 128 + (n & 1);
    const float* ah = a + (size_t)hh * (2 * DD);
    float x0 = wb[2 * lane], x1 = wb[2 * (lane + 32)];
    float s = x0 * ah[lane] + x1 * ah[lane + 32];
    float t = x0 * ah[DD + lane] + x1 * ah[DD + lane + 32];
    for (int off = 16; off > 0; off >>= 1) {
        s += __shfl_down(s, off, 32);
        t += __shfl_down(t, off, 32);
    }
    if (lane == 0) { src[wave] = s; dst[wave] = t; }
}

// ---------------------------------------------------------------------------
// Kernel 3: per-row masked softmax stats (max m, denom l). One wave per row.
// ---------------------------------------------------------------------------
__global__ __launch_bounds__(128) void k_stats(const float* __restrict__ src,
                                               const float* __restrict__ dst,
                                               const int* __restrict__ adj,
                                               float* __restrict__ rowm,
                                               float* __restrict__ rowl) {
    int wave = (int)((blockIdx.x * blockDim.x + threadIdx.x) >> 5);
    int lane = threadIdx.x & 31;
    int bh = wave / NN;
    int i  = wave % NN;
    float si = src[wave];
    const float* dh   = dst + (size_t)bh * NN;
    const int*   arow = adj + (size_t)i * NN;

    float m = -INFINITY;
    for (int j = lane; j < NN; j += 32) {
        if (arow[j] != 0) {
            float e = si + dh[j];
            e = e >= 0.f ? e : LEAKY * e;
            m = fmaxf(m, e);
        }
    }
    for (int off = 16; off > 0; off >>= 1) m = fmaxf(m, __shfl_xor(m, off, 32));

    float lsum = 0.f;
    for (int j = lane; j < NN; j += 32) {
        if (arow[j] != 0) {
            float e = si + dh[j];
            e = e >= 0.f ? e : LEAKY * e;
            lsum += __expf(e - m);
        }
    }
    for (int off = 16; off > 0; off >>= 1) lsum += __shfl_xor(lsum, off, 32);

    if (lane == 0) { rowm[wave] = m; rowl[wave] = lsum; }
}

// ---------------------------------------------------------------------------
// Kernel 4: h_new = softmax(e) @ Wh.
// Block = 4 waves = 64 i-rows of one (b,h). adj ballot-packed to LDS once;
// Wh j-panels (64 rows, 16KB, contiguous in WhP2) streamed to LDS by TDM.
// Each wave: 16x4 A tile of exp-weights built in registers, 4 f32 WMMA
// accumulators cover D=64; B tiles read from LDS as ds_load_b64 pairs.
// ---------------------------------------------------------------------------
__global__ __launch_bounds__(128) void k_attn(const float* __restrict__ WhP2,
                                              const float* __restrict__ src,
                                              const float* __restrict__ dst,
                                              const float* __restrict__ rowm,
                                              const float* __restrict__ rowl,
                                              const int* __restrict__ adj,
                                              float* __restrict__ out) {
    __shared__ __align__(16) float ldsWh[4096];          // 16KB panel
    __shared__ unsigned int maskLDS[64 * 64];            // 16KB adj bitmasks

    int tid  = threadIdx.x;
    int wv   = tid >> 5;
    int lane = tid & 31;
    int half = lane >> 4;
    int l    = lane & 15;

    int bh  = blockIdx.x >> 5;       // 16 (b,h) pairs * 32 row-blocks
    int pb  = blockIdx.x & 31;
    int b   = bh / HH;
    int hh  = bh % HH;
    int ib0 = pb * 64;
    int row0 = ib0 + wv * 16;
    int grow = row0 + l;

    float si = src[(size_t)bh * NN + grow];
    float mi = rowm[(size_t)bh * NN + grow];
    const float* dh = dst + (size_t)bh * NN;

    // ---- ballot-pack adj rows for this wave's 16 i-rows into LDS ----
    for (int r = 0; r < 16; ++r) {
        const int* arow = adj + (size_t)(ib0 + wv * 16 + r) * NN;
        for (int w = 0; w < 64; ++w) {
            unsigned long long bal = __ballot(arow[w * 32 + lane] != 0);
            if (lane == 0) maskLDS[(wv * 16 + r) * 64 + w] = (unsigned int)bal;
        }
    }

    unsigned int ldsOff = (unsigned int)(uintptr_t)(void*)ldsWh; // LDS byte addr
    const float* panel0 = WhP2 + (size_t)bh * (NN / 2) * 128;    // 131072 floats

    v8f c0 = {0.f,0.f,0.f,0.f,0.f,0.f,0.f,0.f};
    v8f c1 = c0, c2 = c0, c3 = c0;

    for (int p = 0; p < NN / 64; ++p) {                 // 32 panels of 64 j
        __syncthreads();                                // panel LDS free
        if (tid < 32)                                   // wave 0 drives the TDM
            tdm_copy_to_lds(ldsOff, panel0 + (size_t)p * 4096, 4096u);
        __builtin_amdgcn_s_wait_tensorcnt(0);
        __syncthreads();                                // panel visible to all

        const float* dp = dh + p * 64;
        const v2f* lwh = (const v2f*)ldsWh;
#pragma unroll
        for (int jw = 0; jw < 2; ++jw) {
            unsigned int mword = maskLDS[(wv * 16 + l) * 64 + (p * 2 + jw)];
#pragma unroll
            for (int jc = 0; jc < 32; jc += 4) {
                int jloc = jw * 32 + jc;                // 0..63 in panel
                int ja   = jloc + 2 * half;             // this lane-half's K pair
                v2f dpr = *(const v2f*)(dp + ja);
                float e0 = si + dpr.x; e0 = e0 >= 0.f ? e0 : LEAKY * e0;
                float e1 = si + dpr.y; e1 = e1 >= 0.f ? e1 : LEAKY * e1;
                unsigned int b0m = (mword >> (jc + 2 * half)) & 1u;
                unsigned int b1m = (mword >> (jc + 2 * half + 1)) & 1u;
                v2f pv;
                pv.x = b0m ? __expf(e0 - mi) : 0.f;     // P[M=l][K=ja]
                pv.y = b1m ? __expf(e1 - mi) : 0.f;     // P[M=l][K=ja+1]

                const v2f* lp = lwh + (size_t)((jloc >> 1) + half) * 64;
                v2f bv0 = lp[l];                        // {B[ja][c], B[ja+1][c]}
                v2f bv1 = lp[16 + l];
                v2f bv2 = lp[32 + l];
                v2f bv3 = lp[48 + l];

                c0 = __builtin_amdgcn_wmma_f32_16x16x4_f32(false, pv, false, bv0, (short)0, c0, false, false);
                c1 = __builtin_amdgcn_wmma_f32_16x16x4_f32(false, pv, false, bv1, (short)0, c1, false, false);
                c2 = __builtin_amdgcn_wmma_f32_16x16x4_f32(false, pv, false, bv2, (short)0, c2, false, false);
                c3 = __builtin_amdgcn_wmma_f32_16x16x4_f32(false, pv, false, bv3, (short)0, c3, false, false);
            }
        }
    }

#pragma unroll
    for (int r = 0; r < 8; ++r) {
        int row   = row0 + r + 8 * half;
        float lv  = rowl[(size_t)bh * NN + row];
        float inv = lv > 0.f ? 1.f / lv : 0.f;          // nan_to_num fallback
        size_t o = (size_t)(b * NN + row) * HD + hh * DD;
        out[o +  0 + l] = c0[r] * inv;
        out[o + 16 + l] = c1[r] * inv;
        out[o + 32 + l] = c2[r] * inv;
        out[o + 48 + l] = c3[r] * inv;
    }
}

// ---------------------------------------------------------------------------
// BatchNorm (training batch stats) + ELU — deterministic 2-stage reduction.
// ---------------------------------------------------------------------------
__global__ __launch_bounds__(256) void k_bnstat(const float* __restrict__ x,
                                                float* __restrict__ p1,
                                                float* __restrict__ p2) {
    int cch = threadIdx.x;                 // channel (coalesced)
    int r0  = blockIdx.x * (ROWS / 64);
    float a = 0.f, bsum = 0.f;
    for (int r = r0; r < r0 + ROWS / 64; ++r) {
        float v = x[(size_t)r * HD + cch];
        a += v; bsum += v * v;
    }
    p1[blockIdx.x * HD + cch] = a;
    p2[blockIdx.x * HD + cch] = bsum;
}

__global__ __launch_bounds__(256) void k_bnred(const float* __restrict__ p1,
                                               const float* __restrict__ p2,
                                               float* __restrict__ s1,
                                               float* __restrict__ s2) {
    int cch = threadIdx.x;
    float a = 0.f, bsum = 0.f;
    for (int g = 0; g < 64; ++g) {         // fixed order -> deterministic
        a    += p1[g * HD + cch];
        bsum += p2[g * HD + cch];
    }
    float mean = a * (1.f / ROWS);
    float var  = bsum * (1.f / ROWS) - mean * mean;
    s1[cch] = mean;
    s2[cch] = rsqrtf(var + EPS_BN);
}

__global__ __launch_bounds__(256) void k_bnelu(float* __restrict__ x,
                                               const float* __restrict__ s1,
                                               const float* __restrict__ s2,
                                               const float* __restrict__ gamma,
                                               const float* __restrict__ beta) {
    size_t idx = (size_t)blockIdx.x * 256 + threadIdx.x;
    int cch = threadIdx.x;                 // HD == blockDim == 256
    float y = (x[idx] - s1[cch]) * s2[cch] * gamma[cch] + beta[cch];
    x[idx] = y > 0.f ? y : (__expf(y) - 1.f);
}

// ---------------------------------------------------------------------------
extern "C" void kernel_launch(void* const* d_in, const int* in_sizes, int n_in,
                              void* d_out, int out_size, void* d_ws, size_t ws_size,
                              hipStream_t stream) {
    const float* h     = (const float*)d_in[0];
    const float* W     = (const float*)d_in[1];
    const float* a     = (const float*)d_in[2];
    const float* gamma = (const float*)d_in[3];
    const float* beta  = (const float*)d_in[4];
    const int*   adj   = (const int*)d_in[5];

    float* ws   = (float*)d_ws;
    float* WhP2 = ws + OFF_WH;
    float* src  = ws + OFF_SRC;
    float* dst  = ws + OFF_DST;
    float* rowm = ws + OFF_M;
    float* rowl = ws + OFF_L;
    float* s1   = ws + OFF_S1;
    float* s2   = ws + OFF_S2;
    float* p1   = ws + OFF_P1;
    float* p2   = ws + OFF_P2;
    float* out  = (float*)d_out;

    k_gemm_wh<<<2048, 128, 0, stream>>>(h, W, WhP2);          // 8192 tiles
    k_srcdst <<<8192, 128, 0, stream>>>(WhP2, a, src, dst);   // 32768 rows
    k_stats  <<<8192, 128, 0, stream>>>(src, dst, adj, rowm, rowl);
    k_attn   <<<512, 128, 0, stream>>>(WhP2, src, dst, rowm, rowl, adj, out);
    k_bnstat <<<64, 256, 0, stream>>>(out, p1, p2);
    k_bnred  <<<1, 256, 0, stream>>>(p1, p2, s1, s2);
    k_bnelu  <<<ROWS * HD / 256, 256, 0, stream>>>(out, s1, s2, gamma, beta);
}